// Flow_43937515438142
// MI455X (gfx1250) — compile-verified
//
#include <hip/hip_runtime.h>
#include <math.h>

#define D_DIM 1024
#define H_DIM 2048

typedef __attribute__((ext_vector_type(2))) float v2f;
typedef __attribute__((ext_vector_type(8))) float v8f;

// ---------------------------------------------------------------------------
// K[a,b] = W2[a,b] * M[b,a],  M = W3 (HxD) @ W1 (DxH)
// One wave32 per 16x16 tile of K. WMMA: D(16x16) = A(16x4) * B(4x16) + C.
// A rows  = b-index (M dim), from W3 rows     (row-major, contiguous in k)
// B cols  = a-index (N dim), from W1 columns  (stride H in k)
// f32 A 16x4 layout: lanes0-15 M=lane, vgpr0=K0 vgpr1=K1; lanes16-31 K2/K3.
// ---------------------------------------------------------------------------
__global__ void k_build_K(const float* __restrict__ W1,
                          const float* __restrict__ W2,
                          const float* __restrict__ W3,
                          float* __restrict__ K) {
  const int lane  = threadIdx.x & 31;
  const int wave  = threadIdx.x >> 5;
  const int tile  = blockIdx.x * 8 + wave;   // 128*128 = 16384 tiles
  const int tileB = tile & 127;
  const int tileA = tile >> 7;
  const int a0 = tileA * 16;
  const int b0 = tileB * 16;
  const int half = lane >> 4;                // selects K pair {0,1} vs {2,3}
  const int lidx = lane & 15;                // M (for A) / N (for B)

  const float* w3row = W3 + (size_t)(b0 + lidx) * D_DIM;
  const float* w1col = W1 + (a0 + lidx);

  v8f c = {};
  for (int k = 0; k < D_DIM; k += 4) {
    const int ka = k + 2 * half;
    v2f a, b;
    a.x = w3row[ka + 0];
    a.y = w3row[ka + 1];
    b.x = w1col[(size_t)(ka + 0) * H_DIM];
    b.y = w1col[(size_t)(ka + 1) * H_DIM];
    c = __builtin_amdgcn_wmma_f32_16x16x4_f32(false, a, false, b,
                                              (short)0, c, false, false);
  }
  // C/D layout: vgpr i holds M=i (lanes0-15) and M=i+8 (lanes16-31), N=lane%16
#pragma unroll
  for (int i = 0; i < 8; ++i) {
    const int bl = i + 8 * half;   // M = local b
    const int al = lidx;           // N = local a
    const size_t idx = (size_t)(a0 + al) * H_DIM + (b0 + bl);
    K[idx] = W2[idx] * c[i];
  }
}

// ---------------------------------------------------------------------------
// Layer kernels: matvecs with LDS staging of the shared input vector.
// ---------------------------------------------------------------------------
__global__ void k_layer1(const float* __restrict__ x,
                         const float* __restrict__ W1,
                         const float* __restrict__ b1,
                         const float* __restrict__ wt, float t,
                         float* __restrict__ h1, float* __restrict__ d1) {
  __shared__ float xs[256];
  const int j = blockIdx.x * 256 + threadIdx.x;   // H outputs
  float acc = b1[j] + t * wt[j];
  for (int i0 = 0; i0 < D_DIM; i0 += 256) {
    __syncthreads();
    xs[threadIdx.x] = x[i0 + threadIdx.x];
    __syncthreads();
#pragma unroll 8
    for (int i = 0; i < 256; ++i)
      acc = fmaf(xs[i], W1[(size_t)(i0 + i) * H_DIM + j], acc);
  }
  const float th = tanhf(acc);
  h1[j] = th;
  d1[j] = 1.0f - th * th;
}

__global__ void k_layer2(const float* __restrict__ h1,
                         const float* __restrict__ W2,
                         const float* __restrict__ b2,
                         float* __restrict__ h2, float* __restrict__ d2) {
  __shared__ float xs[256];
  const int j = blockIdx.x * 256 + threadIdx.x;   // H outputs
  float acc = b2[j];
  for (int i0 = 0; i0 < H_DIM; i0 += 256) {
    __syncthreads();
    xs[threadIdx.x] = h1[i0 + threadIdx.x];
    __syncthreads();
#pragma unroll 8
    for (int i = 0; i < 256; ++i)
      acc = fmaf(xs[i], W2[(size_t)(i0 + i) * H_DIM + j], acc);
  }
  const float th = tanhf(acc);
  h2[j] = th;
  d2[j] = 1.0f - th * th;
}

__global__ void k_layer3(const float* __restrict__ h2,
                         const float* __restrict__ W3,
                         const float* __restrict__ b3,
                         float* __restrict__ v) {
  __shared__ float xs[256];
  const int j = blockIdx.x * 256 + threadIdx.x;   // D outputs
  float acc = b3[j];
  for (int i0 = 0; i0 < H_DIM; i0 += 256) {
    __syncthreads();
    xs[threadIdx.x] = h2[i0 + threadIdx.x];
    __syncthreads();
#pragma unroll 8
    for (int i = 0; i < 256; ++i)
      acc = fmaf(xs[i], W3[(size_t)(i0 + i) * D_DIM + j], acc);
  }
  v[j] = acc;
}

// div partial: per-block p = sum_a d1[a] * (K d2)[a]; deterministic partials.
__global__ void k_div(const float* __restrict__ K,
                      const float* __restrict__ d1,
                      const float* __restrict__ d2,
                      float* __restrict__ part) {
  __shared__ float ds[256];
  __shared__ float red[256];
  const int a = blockIdx.x * 256 + threadIdx.x;   // H rows
  float acc = 0.0f;
  for (int b0 = 0; b0 < H_DIM; b0 += 256) {
    __syncthreads();
    ds[threadIdx.x] = d2[b0 + threadIdx.x];
    __syncthreads();
#pragma unroll 8
    for (int b = 0; b < 256; ++b)
      acc = fmaf(K[(size_t)a * H_DIM + b0 + b], ds[b], acc);
  }
  red[threadIdx.x] = d1[a] * acc;
  __syncthreads();
  for (int s = 128; s > 0; s >>= 1) {
    if (threadIdx.x < s) red[threadIdx.x] += red[threadIdx.x + s];
    __syncthreads();
  }
  if (threadIdx.x == 0) part[blockIdx.x] = red[0];
}

// ---------------------------------------------------------------------------
// Small helpers
// ---------------------------------------------------------------------------
__global__ void k_init(const float* __restrict__ x0, float* __restrict__ xb) {
  const int i = blockIdx.x * 256 + threadIdx.x;
  if (i < D_DIM) xb[i] = x0[i];
}

__global__ void k_logq0(const float* __restrict__ x0, float* __restrict__ scal) {
  __shared__ float red[256];
  float acc = 0.0f;
  for (int i = threadIdx.x; i < D_DIM; i += 256) {
    const float v = x0[i];
    acc += v * v;
  }
  red[threadIdx.x] = acc;
  __syncthreads();
  for (int s = 128; s > 0; s >>= 1) {
    if (threadIdx.x < s) red[threadIdx.x] += red[threadIdx.x + s];
    __syncthreads();
  }
  if (threadIdx.x == 0)
    scal[0] = -0.5f * red[0]
            - 0.5f * (float)D_DIM * logf(2.0f * 3.14159265358979323846f);
}

__global__ void k_axpy(const float* __restrict__ x, const float* __restrict__ v,
                       float s, float* __restrict__ out) {
  const int i = blockIdx.x * 256 + threadIdx.x;
  if (i < D_DIM) out[i] = x[i] + s * v[i];
}

__global__ void k_update(float* __restrict__ x,
                         const float* __restrict__ v1, const float* __restrict__ v2,
                         const float* __restrict__ v3, const float* __restrict__ v4,
                         const float* __restrict__ divP,  // [4][8] partials
                         float* __restrict__ scal, float dt) {
  const int i = blockIdx.x * 256 + threadIdx.x;
  const float s = dt / 6.0f;
  if (i < D_DIM)
    x[i] += s * (v1[i] + 2.0f * v2[i] + 2.0f * v3[i] + v4[i]);
  if (i == 0) {
    float dl[4];
    for (int st = 0; st < 4; ++st) {
      float d = 0.0f;
      for (int k = 0; k < 8; ++k) d += divP[st * 8 + k];
      dl[st] = d;
    }
    // dlogq/dt = -div
    scal[0] -= s * (dl[0] + 2.0f * dl[1] + 2.0f * dl[2] + dl[3]);
  }
}

__global__ void k_finalize(const float* __restrict__ x,
                           const float* __restrict__ scal,
                           float* __restrict__ out) {
  const int i = blockIdx.x * 256 + threadIdx.x;
  if (i < D_DIM) out[i] = x[i];
  if (i == D_DIM) out[D_DIM] = scal[0];
}

// ---------------------------------------------------------------------------
extern "C" void kernel_launch(void* const* d_in, const int* in_sizes, int n_in,
                              void* d_out, int out_size, void* d_ws, size_t ws_size,
                              hipStream_t stream) {
  const float* x0 = (const float*)d_in[0];
  const float* W1 = (const float*)d_in[1];
  const float* b1 = (const float*)d_in[2];
  const float* wt = (const float*)d_in[3];
  const float* W2 = (const float*)d_in[4];
  const float* b2 = (const float*)d_in[5];
  const float* W3 = (const float*)d_in[6];
  const float* b3 = (const float*)d_in[7];
  float* out = (float*)d_out;

  float* ws = (float*)d_ws;
  float* K    = ws;                                  // H*H
  float* xb   = K + (size_t)H_DIM * H_DIM;           // D (current state)
  float* xt   = xb + D_DIM;                          // D (stage input)
  float* h1   = xt + D_DIM;                          // H
  float* d1   = h1 + H_DIM;                          // H
  float* h2   = d1 + H_DIM;                          // H
  float* d2   = h2 + H_DIM;                          // H
  float* v1   = d2 + H_DIM;                          // D x4
  float* v2   = v1 + D_DIM;
  float* v3   = v2 + D_DIM;
  float* v4   = v3 + D_DIM;
  float* scal = v4 + D_DIM;                          // [0] = logq
  float* divP = scal + 8;                            // 4 stages * 8 partials

  // K = W2 .* (W3@W1)^T  -- the only GEMM; WMMA f32 16x16x4
  k_build_K<<<2048, 256, 0, stream>>>(W1, W2, W3, K);
  k_init<<<4, 256, 0, stream>>>(x0, xb);
  k_logq0<<<1, 256, 0, stream>>>(x0, scal);

  const float dt = 0.5f;   // (T1-T0)/N_STEPS
  float* vs[4] = {v1, v2, v3, v4};

  for (int step = 0; step < 2; ++step) {
    const float t = (float)step * dt;
    for (int st = 0; st < 4; ++st) {
      const float ts = t + ((st == 0) ? 0.0f : (st == 3) ? dt : 0.5f * dt);
      const float* xin = (st == 0) ? xb : xt;
      k_layer1<<<H_DIM / 256, 256, 0, stream>>>(xin, W1, b1, wt, ts, h1, d1);
      k_layer2<<<H_DIM / 256, 256, 0, stream>>>(h1, W2, b2, h2, d2);
      k_div   <<<H_DIM / 256, 256, 0, stream>>>(K, d1, d2, divP + st * 8);
      k_layer3<<<D_DIM / 256, 256, 0, stream>>>(h2, W3, b3, vs[st]);
      if (st < 3) {
        const float sc = (st == 2) ? dt : 0.5f * dt;
        k_axpy<<<D_DIM / 256, 256, 0, stream>>>(xb, vs[st], sc, xt);
      }
    }
    k_update<<<D_DIM / 256, 256, 0, stream>>>(xb, v1, v2, v3, v4, divP, scal, dt);
  }
  k_finalize<<<(D_DIM + 256) / 256 + 1, 256, 0, stream>>>(xb, scal, out);
}